// RNN_67628555043381
// MI455X (gfx1250) — compile-verified
//
#include <hip/hip_runtime.h>
#include <hip/hip_bf16.h>

typedef __attribute__((ext_vector_type(2))) float v2f;
typedef __attribute__((ext_vector_type(4))) float v4f;
typedef __attribute__((ext_vector_type(8))) float v8f;

#define H_DIM 4096
#define TILE_M 16
#define CK 256             // K-chunk staged in LDS per iteration
#define LDS_STRIDE 260     // 256 + 4 pad: (4*row + col) mod 64 -> conflict-free
#define BLOCK 256          // 8 waves
#define F4PT 4             // float4 loads per thread per chunk (16*256/4/256)

// ---------------------------------------------------------------------------
// y[m0..m0+15] = b[m] + sum_k W[m,k] * v[k]   via V_WMMA_F32_16X16X4_F32
// Software-pipelined: global->regs for chunk n+1 issued before WMMA on chunk n,
// regs->LDS commit (and its loadcnt wait) after the WMMA block.
// gridDim.y selects between two (W,v,b,y) jobs (both GEMVs of a GRU layer).
// ---------------------------------------------------------------------------
__global__ __launch_bounds__(BLOCK) void gemv16_wmma_kernel(
    const float* __restrict__ W0, const float* __restrict__ v0,
    const float* __restrict__ b0, float* __restrict__ y0,
    const float* __restrict__ W1, const float* __restrict__ v1,
    const float* __restrict__ b1, float* __restrict__ y1,
    int K)
{
    __shared__ float lds[2][TILE_M * LDS_STRIDE];
    __shared__ float red[TILE_M];

    const float* __restrict__ W;
    const float* __restrict__ v;
    const float* __restrict__ b;
    float* __restrict__ y;
    if (blockIdx.y == 0) { W = W0; v = v0; b = b0; y = y0; }
    else                 { W = W1; v = v1; b = b1; y = y1; }

    const int t       = threadIdx.x;
    const int lane    = t & 31;
    const int wave    = t >> 5;
    const int row     = lane & 15;        // A-matrix M within tile
    const int halfsel = lane >> 4;        // 0: K+0/K+1, 1: K+2/K+3
    const int m0      = blockIdx.x * TILE_M;

    if (t < TILE_M) red[t] = 0.0f;

    // Per-thread staging geometry (constant across chunks): 1024 float4 per
    // chunk, 4 per thread. Precompute global row pointers + LDS offsets once.
    const float* gp[F4PT];
    int          lo[F4PT];
#pragma unroll
    for (int j = 0; j < F4PT; ++j) {
        int i = t + BLOCK * j;            // float4 index 0..1023
        int r = i >> 6;                   // row 0..15 (64 float4 per row)
        int c = (i & 63) << 2;            // col 0..252
        gp[j] = W + (size_t)(m0 + r) * K + c;
        lo[j] = r * LDS_STRIDE + c;
    }

    auto load_chunk = [&](int k0, v4f* regs) {
#pragma unroll
        for (int j = 0; j < F4PT; ++j)
            regs[j] = *(const v4f*)(gp[j] + k0);
    };
    auto commit_chunk = [&](int buf, const v4f* regs) {
#pragma unroll
        for (int j = 0; j < F4PT; ++j)
            *(v4f*)&lds[buf][lo[j]] = regs[j];
    };

    v8f acc = {};   // 16x16 f32 accumulator (all 16 columns identical)
    v4f regs[F4PT];

    load_chunk(0, regs);
    commit_chunk(0, regs);
    __syncthreads();

    const int nchunks = K / CK;
    for (int ch = 0; ch < nchunks; ++ch) {
        const int  buf  = ch & 1;
        const bool more = (ch + 1 < nchunks);

        if (more) load_chunk((ch + 1) * CK, regs);   // global loads in flight

        // Each wave consumes 32 of the 256 staged columns -> 8 WMMAs.
#pragma unroll
        for (int kk = 0; kk < 8; ++kk) {
            int kloc = wave * 32 + kk * 4 + 2 * halfsel;
            v2f a = *(const v2f*)&lds[buf][row * LDS_STRIDE + kloc];
            v2f bf = *(const v2f*)(v + ch * CK + kloc);  // broadcast vector
            acc = __builtin_amdgcn_wmma_f32_16x16x4_f32(
                false, a, false, bf, (short)0, acc, false, false);
        }

        if (more) commit_chunk(buf ^ 1, regs);       // loadcnt wait lands here
        __syncthreads();
    }

    // Column 0 of D: lane 0 holds rows 0..7, lane 16 holds rows 8..15.
    if (row == 0) {
        int base = halfsel * 8;
#pragma unroll
        for (int i = 0; i < 8; ++i) atomicAdd(&red[base + i], acc[i]);
    }
    __syncthreads();
    if (t < TILE_M) y[m0 + t] = red[t] + b[m0 + t];
}

// ---------------------------------------------------------------------------
__global__ void embed_kernel(const int* __restrict__ inp,
                             const float* __restrict__ emb,
                             float* __restrict__ x)
{
    int i = blockIdx.x * blockDim.x + threadIdx.x;
    int tok = inp[0];
    if (i < H_DIM) x[i] = emb[(size_t)tok * H_DIM + i];
}

__global__ void gru_combine_kernel(const float* __restrict__ gi,
                                   const float* __restrict__ gh,
                                   const float* __restrict__ h_in,
                                   float* __restrict__ h_out)
{
    int i = blockIdx.x * blockDim.x + threadIdx.x;
    if (i >= H_DIM) return;
    float r = 1.0f / (1.0f + __expf(-(gi[i] + gh[i])));
    float z = 1.0f / (1.0f + __expf(-(gi[H_DIM + i] + gh[H_DIM + i])));
    float n = tanhf(gi[2 * H_DIM + i] + r * gh[2 * H_DIM + i]);
    h_out[i] = (1.0f - z) * n + z * h_in[i];
}

// ---------------------------------------------------------------------------
extern "C" void kernel_launch(void* const* d_in, const int* in_sizes, int n_in,
                              void* d_out, int out_size, void* d_ws, size_t ws_size,
                              hipStream_t stream)
{
    const int*   inp    = (const int*)  d_in[0];
    const float* hidden = (const float*)d_in[1];   // [2,1,4096]
    const float* emb    = (const float*)d_in[2];   // [256,4096]
    const float* W_ih0  = (const float*)d_in[3];
    const float* W_hh0  = (const float*)d_in[4];
    const float* b_ih0  = (const float*)d_in[5];
    const float* b_hh0  = (const float*)d_in[6];
    const float* W_ih1  = (const float*)d_in[7];
    const float* W_hh1  = (const float*)d_in[8];
    const float* b_ih1  = (const float*)d_in[9];
    const float* b_hh1  = (const float*)d_in[10];
    const float* W_dec  = (const float*)d_in[11];
    const float* b_dec  = (const float*)d_in[12];

    float* out    = (float*)d_out;           // [0:256) logits
    float* h0_out = out + 256;               // [256:4352) new h layer0
    float* h1_out = out + 256 + H_DIM;       // [4352:8448) new h layer1

    float* ws = (float*)d_ws;
    float* x0 = ws;                          // 4096
    float* gi = ws + H_DIM;                  // 12288
    float* gh = ws + H_DIM + 3 * H_DIM;      // 12288

    const float* h0_in = hidden;
    const float* h1_in = hidden + H_DIM;

    // 1) encoder embedding lookup
    embed_kernel<<<dim3(H_DIM / 256), dim3(256), 0, stream>>>(inp, emb, x0);

    // 2) layer 0 gates: gi = W_ih0 @ x0 + b_ih0 ; gh = W_hh0 @ h0 + b_hh0
    gemv16_wmma_kernel<<<dim3(3 * H_DIM / TILE_M, 2), dim3(BLOCK), 0, stream>>>(
        W_ih0, x0, b_ih0, gi,
        W_hh0, h0_in, b_hh0, gh, H_DIM);

    // 3) layer 0 nonlinearity -> h0_out
    gru_combine_kernel<<<dim3(H_DIM / 256), dim3(256), 0, stream>>>(gi, gh, h0_in, h0_out);

    // 4) layer 1 gates (input is h0_out)
    gemv16_wmma_kernel<<<dim3(3 * H_DIM / TILE_M, 2), dim3(BLOCK), 0, stream>>>(
        W_ih1, h0_out, b_ih1, gi,
        W_hh1, h1_in, b_hh1, gh, H_DIM);

    // 5) layer 1 nonlinearity -> h1_out
    gru_combine_kernel<<<dim3(H_DIM / 256), dim3(256), 0, stream>>>(gi, gh, h1_in, h1_out);

    // 6) decoder: out = W_dec @ h1_out + b_dec  (M=256 -> 16 tiles)
    gemv16_wmma_kernel<<<dim3(256 / TILE_M, 1), dim3(BLOCK), 0, stream>>>(
        W_dec, h1_out, b_dec, out,
        W_dec, h1_out, b_dec, out, H_DIM);

    (void)in_sizes; (void)n_in; (void)out_size; (void)ws_size;
}